// APPNP_51573967290853
// MI455X (gfx1250) — compile-verified
//
#include <hip/hip_runtime.h>
#include <hip/hip_bf16.h>

// ---------------------------------------------------------------------------
// APPNP on MI455X (gfx1250, wave32):
//   h0 = relu(feats@W1+b1)@W2+b2   -> fused WMMA kernel (V_WMMA_F32_16X16X4_F32)
//   10x: agg = scatter_add(gather(h*norm_out)); h = 0.9*agg*norm_in + 0.1*h0
// ---------------------------------------------------------------------------

typedef float v2f __attribute__((ext_vector_type(2)));
typedef float v8f __attribute__((ext_vector_type(8)));

#define IN_DIM  512
#define HID_DIM 256
#define OUT_DIM 48
#define ALPHA   0.1f

__device__ __forceinline__ v8f wmma_f32_16x16x4(v2f a, v2f b, v8f c) {
    // 8-arg pattern: (neg_a, A, neg_b, B, c_mod, C, reuse_a, reuse_b)
    return __builtin_amdgcn_wmma_f32_16x16x4_f32(false, a, false, b, (short)0, c,
                                                 false, false);
}

// LDS row strides (padded, multiples of 4 for float4 aliasing)
#define AS_STRIDE 516   // 512 + 4
#define HS_STRIDE 264   // 256 + 8

// One block = 16 rows of the node matrix. 256 threads = 8 waves.
// Phase 1: H[16][256] = relu(A[16][512] @ W1 + b1) kept in LDS.
// Phase 2: OUT[16][48] = H @ W2 + b2, written to h0 (ws) and hout (d_out).
__global__ void __launch_bounds__(256)
mlp_fused_wmma_kernel(const float* __restrict__ feats,
                      const float* __restrict__ W1, const float* __restrict__ b1,
                      const float* __restrict__ W2, const float* __restrict__ b2,
                      float* __restrict__ h0, float* __restrict__ hout, int N) {
    __shared__ float As[16 * AS_STRIDE];
    __shared__ float Hs[16 * HS_STRIDE];

    const int rowBase = blockIdx.x * 16;
    const int tid  = threadIdx.x;
    const int lane = tid & 31;
    const int wave = tid >> 5;
    const int half = lane >> 4;   // 0: lanes 0-15, 1: lanes 16-31
    const int m16  = lane & 15;

    // ---- stage A strip (16 x 512 f32) into LDS, float4-coalesced ----
    {
        const float4* A4  = reinterpret_cast<const float4*>(feats);
        float4*       As4 = reinterpret_cast<float4*>(As);
        for (int idx = tid; idx < 16 * 128; idx += 256) {
            const int r  = idx >> 7;       // row 0..15
            const int c4 = idx & 127;      // float4 column 0..127
            float4 v;
            if (rowBase + r < N) {
                v = A4[(size_t)(rowBase + r) * (IN_DIM / 4) + c4];
            } else {
                v = make_float4(0.f, 0.f, 0.f, 0.f);
            }
            As4[r * (AS_STRIDE / 4) + c4] = v;
        }
    }
    __syncthreads();

    // ---- phase 1: each wave owns cols [16w,16w+16) and [16w+128,16w+144) ----
    {
        const int c0 = wave * 16 + m16;
        const int c1 = wave * 16 + 128 + m16;
        v8f acc0 = {};
        v8f acc1 = {};
#pragma unroll 4
        for (int k0 = 0; k0 < IN_DIM; k0 += 4) {
            const int ka = k0 + 2 * half;     // this half-wave covers K=ka, ka+1
            v2f a;
            a.x = As[m16 * AS_STRIDE + ka];
            a.y = As[m16 * AS_STRIDE + ka + 1];
            v2f bA, bB;
            bA.x = W1[(size_t)ka * HID_DIM + c0];
            bA.y = W1[(size_t)(ka + 1) * HID_DIM + c0];
            bB.x = W1[(size_t)ka * HID_DIM + c1];
            bB.y = W1[(size_t)(ka + 1) * HID_DIM + c1];
            acc0 = wmma_f32_16x16x4(a, bA, acc0);
            acc1 = wmma_f32_16x16x4(a, bB, acc1);
        }
        const float bias0 = b1[c0];
        const float bias1 = b1[c1];
#pragma unroll
        for (int r = 0; r < 8; ++r) {
            const int row = r + 8 * half;     // D layout: VGPR r = row r (+8 hi half)
            Hs[row * HS_STRIDE + c0] = fmaxf(acc0[r] + bias0, 0.f);
            Hs[row * HS_STRIDE + c1] = fmaxf(acc1[r] + bias1, 0.f);
        }
    }
    __syncthreads();

    // ---- phase 2: waves 0..2 compute the 3 output col-tiles (48 = 3*16) ----
    if (wave < 3) {
        const int c = wave * 16 + m16;
        v8f acc = {};
#pragma unroll 4
        for (int k0 = 0; k0 < HID_DIM; k0 += 4) {
            const int ka = k0 + 2 * half;
            v2f a;
            a.x = Hs[m16 * HS_STRIDE + ka];
            a.y = Hs[m16 * HS_STRIDE + ka + 1];
            v2f b;
            b.x = W2[(size_t)ka * OUT_DIM + c];
            b.y = W2[(size_t)(ka + 1) * OUT_DIM + c];
            acc = wmma_f32_16x16x4(a, b, acc);
        }
        const float bias = b2[c];
#pragma unroll
        for (int r = 0; r < 8; ++r) {
            const int row = rowBase + r + 8 * half;
            if (row < N) {
                const float v = acc[r] + bias;
                h0[(size_t)row * OUT_DIM + c]   = v;
                hout[(size_t)row * OUT_DIM + c] = v;
            }
        }
    }
}

// ---------------------------------------------------------------------------

__global__ void fill_zero_kernel(float* __restrict__ p, long long n) {
    long long i = (long long)blockIdx.x * blockDim.x + threadIdx.x;
    if (i < n) p[i] = 0.f;
}

__global__ void degree_kernel(const int* __restrict__ src, const int* __restrict__ dst,
                              float* __restrict__ deg_out, float* __restrict__ deg_in, int E) {
    int e = blockIdx.x * blockDim.x + threadIdx.x;
    if (e < E) {
        atomicAdd(&deg_out[src[e]], 1.f);
        atomicAdd(&deg_in[dst[e]], 1.f);
    }
}

// in place: deg -> rsqrt(max(deg,1))
__global__ void norm_kernel(float* __restrict__ deg, long long n) {
    long long i = (long long)blockIdx.x * blockDim.x + threadIdx.x;
    if (i < n) deg[i] = rsqrtf(fmaxf(deg[i], 1.f));
}

// agg[dst[e]][ch] += h[src[e]][ch] * norm_out[src[e]]
__global__ void prop_edge_kernel(const int* __restrict__ src, const int* __restrict__ dst,
                                 const float* __restrict__ h, const float* __restrict__ nrm_out,
                                 float* __restrict__ agg, long long total) {
    long long i = (long long)blockIdx.x * blockDim.x + threadIdx.x;
    if (i >= total) return;
    const int e  = (int)(i / OUT_DIM);
    const int ch = (int)(i - (long long)e * OUT_DIM);
    const int s = src[e];
    const int d = dst[e];
    const float v = h[(size_t)s * OUT_DIM + ch] * nrm_out[s];
    atomicAdd(&agg[(size_t)d * OUT_DIM + ch], v);
}

// out = (1-alpha) * agg * norm_in + alpha * h0
__global__ void combine_kernel(const float* __restrict__ agg, const float* __restrict__ h0,
                               const float* __restrict__ nrm_in, float* __restrict__ out,
                               long long total) {
    long long i = (long long)blockIdx.x * blockDim.x + threadIdx.x;
    if (i >= total) return;
    const int n = (int)(i / OUT_DIM);
    out[i] = (1.f - ALPHA) * agg[i] * nrm_in[n] + ALPHA * h0[i];
}

// ---------------------------------------------------------------------------

extern "C" void kernel_launch(void* const* d_in, const int* in_sizes, int n_in,
                              void* d_out, int out_size, void* d_ws, size_t ws_size,
                              hipStream_t stream) {
    // setup_inputs order: feats, src, dst, W1, b1, W2, b2
    const float* feats = (const float*)d_in[0];
    const int*   src   = (const int*)d_in[1];
    const int*   dst   = (const int*)d_in[2];
    const float* W1    = (const float*)d_in[3];
    const float* b1    = (const float*)d_in[4];
    const float* W2    = (const float*)d_in[5];
    const float* b2    = (const float*)d_in[6];

    const int N = in_sizes[0] / IN_DIM;
    const int E = in_sizes[1];

    float* out = (float*)d_out;

    // workspace layout (floats)
    float* h0      = (float*)d_ws;
    float* hbuf    = h0 + (size_t)N * OUT_DIM;
    float* agg     = hbuf + (size_t)N * OUT_DIM;
    float* nrm_out = agg + (size_t)N * OUT_DIM;
    float* nrm_in  = nrm_out + N;   // contiguous with nrm_out

    const int T = 256;
    const long long featTotal = (long long)N * OUT_DIM;
    const long long edgeTotal = (long long)E * OUT_DIM;

    // degrees -> norms (nrm_out/nrm_in are contiguous: one fill, one norm pass)
    fill_zero_kernel<<<(unsigned)((2LL * N + T - 1) / T), T, 0, stream>>>(nrm_out, 2LL * N);
    degree_kernel<<<(E + T - 1) / T, T, 0, stream>>>(src, dst, nrm_out, nrm_in, E);
    norm_kernel<<<(unsigned)((2LL * N + T - 1) / T), T, 0, stream>>>(nrm_out, 2LL * N);

    // fused MLP: h0 -> ws, initial h -> d_out
    mlp_fused_wmma_kernel<<<(N + 15) / 16, 256, 0, stream>>>(feats, W1, b1, W2, b2,
                                                             h0, out, N);

    // K=10 propagation rounds; ping-pong so round 10 (even) lands in d_out
    for (int it = 1; it <= 10; ++it) {
        const float* cur = (it & 1) ? out  : hbuf;
        float*       nxt = (it & 1) ? hbuf : out;
        fill_zero_kernel<<<(unsigned)((featTotal + T - 1) / T), T, 0, stream>>>(agg, featTotal);
        prop_edge_kernel<<<(unsigned)((edgeTotal + T - 1) / T), T, 0, stream>>>(
            src, dst, cur, nrm_out, agg, edgeTotal);
        combine_kernel<<<(unsigned)((featTotal + T - 1) / T), T, 0, stream>>>(
            agg, h0, nrm_in, nxt, featTotal);
    }
}